// StructureModule_62466004353656
// MI455X (gfx1250) — compile-verified
//
#include <hip/hip_runtime.h>
#include <math.h>

typedef __attribute__((ext_vector_type(2))) float v2f;
typedef __attribute__((ext_vector_type(8))) float v8f;

#define NRES 512
#define CSD  384
#define CZD  128
#define HN   12
#define CHN  16
#define PQN  4
#define PVN  8
#define OUTDIM 2112   // H*(CZ+CH+PV*4)
#define INF_ 100000.0f
#define EPS_ 1e-8f

__device__ __forceinline__ v8f wmma4(v2f a, v2f b, v8f c) {
  // D = A(16x4,f32) * B(4x16,f32) + C(16x16,f32)
  return __builtin_amdgcn_wmma_f32_16x16x4_f32(false, a, false, b, (short)0, c, false, false);
}

// ---------------------------------------------------------------------------
// K1: per-residue projections.  Builds:
//   qc[h][i][32] = [ q*sqrt(1/48) (16) | q_pts*hw_h (12) | 0 pad (4) ]
//   kc[h][i][32] = [ k (16)            | k_pts (12)      | 0 pad (4) ]
//   vc[h][i][48] = [ v (16) | v_pts (24) | 0 pad (8) ]
//   rb[h][i] = -0.5*hw_h*sum|q_pts|^2 ; cb[h][i] = -0.5*hw_h*sum|k_pts|^2
// ---------------------------------------------------------------------------
__global__ void k1_proj(const float* __restrict__ s, const float* __restrict__ rot,
                        const float* __restrict__ trans,
                        const float* __restrict__ Wq,  const float* __restrict__ bq,
                        const float* __restrict__ Wkv, const float* __restrict__ bkv,
                        const float* __restrict__ Wqp, const float* __restrict__ bqp,
                        const float* __restrict__ Wkvp,const float* __restrict__ bkvp,
                        const float* __restrict__ head_w,
                        float* __restrict__ qc, float* __restrict__ kc,
                        float* __restrict__ vc, float* __restrict__ rb,
                        float* __restrict__ cb) {
  __shared__ float s_sh[CSD];
  __shared__ float proj[1152];
  __shared__ float hw_sh[HN];
  __shared__ float rbacc[HN];
  __shared__ float cbacc[HN];
  const int i = blockIdx.x;
  const int t = threadIdx.x;
  for (int u = t; u < CSD; u += 256) s_sh[u] = s[i*CSD + u];
  if (t < HN) {
    float w = head_w[t];
    hw_sh[t] = log1pf(__expf(w)) * 0.13608276348795434f;   // softplus * sqrt(1/54)
    rbacc[t] = 0.f; cbacc[t] = 0.f;
  }
  __syncthreads();
  // flat projected columns: [0,192) q | [192,576) kv | [576,720) qp | [720,1152) kvp
  for (int col = t; col < 1152; col += 256) {
    const float* W; float bias; int stride, c;
    if (col < 192)      { W = Wq;   c = col;       stride = 192; bias = bq[c];  }
    else if (col < 576) { W = Wkv;  c = col - 192; stride = 384; bias = bkv[c]; }
    else if (col < 720) { W = Wqp;  c = col - 576; stride = 144; bias = bqp[c]; }
    else                { W = Wkvp; c = col - 720; stride = 432; bias = bkvp[c];}
    float acc = bias;
    for (int r = 0; r < CSD; ++r) acc = fmaf(s_sh[r], W[r*stride + c], acc);
    proj[col] = acc;
  }
  __syncthreads();
  const float S1 = 0.14433756729740643f;  // sqrt(1/(3*CH))
  float R[9], T[3];
  #pragma unroll
  for (int u = 0; u < 9; ++u) R[u] = rot[i*9 + u];
  T[0] = trans[i*3]; T[1] = trans[i*3+1]; T[2] = trans[i*3+2];
  for (int u = t; u < 192; u += 256) {            // q
    int h = u >> 4, c = u & 15;
    qc[(h*NRES + i)*32 + c] = proj[u] * S1;
  }
  for (int u = t; u < 384; u += 256) {            // k / v
    int h = u >> 5, cc = u & 31;
    float val = proj[192 + u];
    if (cc < 16) kc[(h*NRES + i)*32 + cc] = val;
    else         vc[(h*NRES + i)*48 + (cc - 16)] = val;
  }
  for (int u = t; u < 48; u += 256) {             // q_pts (H*PQ)
    int h = u >> 2, p = u & 3;
    float p0 = proj[576 + 0*48 + u], p1 = proj[576 + 1*48 + u], p2 = proj[576 + 2*48 + u];
    float r0 = R[0]*p0 + R[1]*p1 + R[2]*p2 + T[0];
    float r1 = R[3]*p0 + R[4]*p1 + R[5]*p2 + T[1];
    float r2 = R[6]*p0 + R[7]*p1 + R[8]*p2 + T[2];
    float hw = hw_sh[h];
    float* dst = &qc[(h*NRES + i)*32 + 16 + p*3];
    dst[0] = r0*hw; dst[1] = r1*hw; dst[2] = r2*hw;
    atomicAdd(&rbacc[h], r0*r0 + r1*r1 + r2*r2);
  }
  for (int u = t; u < 144; u += 256) {            // kv_pts (H*(PQ+PV))
    int h = u / 12, pp = u % 12;
    float p0 = proj[720 + 0*144 + u], p1 = proj[720 + 1*144 + u], p2 = proj[720 + 2*144 + u];
    float r0 = R[0]*p0 + R[1]*p1 + R[2]*p2 + T[0];
    float r1 = R[3]*p0 + R[4]*p1 + R[5]*p2 + T[1];
    float r2 = R[6]*p0 + R[7]*p1 + R[8]*p2 + T[2];
    if (pp < PQN) {
      float* dst = &kc[(h*NRES + i)*32 + 16 + pp*3];
      dst[0] = r0; dst[1] = r1; dst[2] = r2;
      atomicAdd(&cbacc[h], r0*r0 + r1*r1 + r2*r2);
    } else {
      float* dst = &vc[(h*NRES + i)*48 + 16 + (pp - PQN)*3];
      dst[0] = r0; dst[1] = r1; dst[2] = r2;
    }
  }
  for (int u = t; u < HN; u += 256) {             // zero padding lanes
    for (int kk = 28; kk < 32; ++kk) { qc[(u*NRES+i)*32+kk] = 0.f; kc[(u*NRES+i)*32+kk] = 0.f; }
    for (int kk = 40; kk < 48; ++kk) vc[(u*NRES+i)*48+kk] = 0.f;
  }
  __syncthreads();
  if (t < HN) {
    rb[t*NRES + i] = -0.5f * hw_sh[t] * rbacc[t];
    cb[t*NRES + i] = -0.5f * hw_sh[t] * cbacc[t];
  }
}

// ---------------------------------------------------------------------------
// K2: b_pair = z @ Wb + bb.  (N*N x 128)@(128 x 12).  One 16-row tile per wave.
// ---------------------------------------------------------------------------
__global__ void k2_bpair(const float* __restrict__ z, const float* __restrict__ Wb,
                         const float* __restrict__ bb, float* __restrict__ bpair) {
  __shared__ float wb_sh[128*16];
  __shared__ float bb_sh[16];
  const int t = threadIdx.x;
  for (int u = t; u < 2048; u += 256) {
    int k = u >> 4, c = u & 15;
    wb_sh[u] = (c < HN) ? Wb[k*HN + c] : 0.f;
  }
  if (t < 16) bb_sh[t] = (t < HN) ? bb[t] : 0.f;
  __syncthreads();
  const int wave = t >> 5, lane = t & 31;
  const int rowbase = (blockIdx.x*8 + wave) * 16;
  const int rc = lane & 15, hi = lane >> 4, koff = hi*2;
  v8f c = {};
  for (int ks = 0; ks < 32; ++ks) {
    int k0 = ks*4 + koff;
    v2f a, b;
    a.x = z[(size_t)(rowbase + rc)*CZD + k0];
    a.y = z[(size_t)(rowbase + rc)*CZD + k0 + 1];
    b.x = wb_sh[k0*16 + rc];
    b.y = wb_sh[(k0+1)*16 + rc];
    c = wmma4(a, b, c);
  }
  if (rc < HN) {
    #pragma unroll
    for (int r = 0; r < 8; ++r) {
      int m = r + hi*8;
      bpair[(size_t)(rowbase + m)*HN + rc] = c[r] + bb_sh[rc];
    }
  }
}

// ---------------------------------------------------------------------------
// K3: attention logits via K=32 WMMA + biases + mask, then row softmax.
// One block per (head, 16-row i-tile); 8 waves x 4 j-tiles each cover j=0..511.
// ---------------------------------------------------------------------------
__global__ void k3_attn(const float* __restrict__ qc, const float* __restrict__ kc,
                        const float* __restrict__ rb, const float* __restrict__ cb,
                        const float* __restrict__ bpair, const float* __restrict__ mask,
                        float* __restrict__ attn) {
  __shared__ float qsh[16*36];
  __shared__ float logits[16*512];
  __shared__ float rbsh[16], mrow[16];
  __shared__ float cbsh[512], msh[512];
  __shared__ float red[256];
  const int t = threadIdx.x;
  const int h = blockIdx.x >> 5;
  const int it = blockIdx.x & 31;
  for (int u = t; u < 512; u += 256) {
    int r = u >> 5, kk = u & 31;
    qsh[r*36 + kk] = qc[(h*NRES + it*16 + r)*32 + kk];
  }
  for (int u = t; u < 512; u += 256) { cbsh[u] = cb[h*NRES + u]; msh[u] = mask[u]; }
  if (t < 16) { rbsh[t] = rb[h*NRES + it*16 + t]; mrow[t] = mask[it*16 + t]; }
  __syncthreads();
  const int wave = t >> 5, lane = t & 31;
  const int rc = lane & 15, hi = lane >> 4, koff = hi*2;
  const float S13 = 0.5773502691896258f;  // sqrt(1/3)
  for (int jj = 0; jj < 4; ++jj) {
    const int jt = wave*4 + jj;
    v8f c = {};
    for (int ks = 0; ks < 8; ++ks) {
      int k0 = ks*4 + koff;
      v2f a, b;
      a.x = qsh[rc*36 + k0];
      a.y = qsh[rc*36 + k0 + 1];
      b.x = kc[(h*NRES + jt*16 + rc)*32 + k0];
      b.y = kc[(h*NRES + jt*16 + rc)*32 + k0 + 1];
      c = wmma4(a, b, c);
    }
    #pragma unroll
    for (int r = 0; r < 8; ++r) {
      int m = r + hi*8;
      int ig = it*16 + m, j = jt*16 + rc;
      float val = c[r] + rbsh[m] + cbsh[j]
                + S13 * bpair[((size_t)ig*NRES + j)*HN + h]
                + INF_ * (mrow[m]*msh[j] - 1.0f);
      logits[m*512 + j] = val;
    }
  }
  __syncthreads();
  const int row = t >> 4, tr = t & 15;
  float* lrow = &logits[row*512 + tr*32];
  float mx = -3.0e38f;
  for (int q = 0; q < 32; ++q) mx = fmaxf(mx, lrow[q]);
  red[row*16 + tr] = mx;
  __syncthreads();
  for (int q = 0; q < 16; ++q) mx = fmaxf(mx, red[row*16 + q]);
  float ssum = 0.f;
  for (int q = 0; q < 32; ++q) { float e = __expf(lrow[q] - mx); lrow[q] = e; ssum += e; }
  __syncthreads();
  red[row*16 + tr] = ssum;
  __syncthreads();
  float tot = 0.f;
  for (int q = 0; q < 16; ++q) tot += red[row*16 + q];
  const float inv = 1.0f / tot;
  float* orow = &attn[((size_t)h*NRES + it*16 + row)*NRES + tr*32];
  for (int q = 0; q < 32; ++q) orow[q] = lrow[q] * inv;
}

// ---------------------------------------------------------------------------
// K4a: [o | o_pt_raw] = a @ vc.  Per wave: one (h, i-tile, n-tile) job, K=512.
// ---------------------------------------------------------------------------
__global__ void k4a_ov(const float* __restrict__ attn, const float* __restrict__ vc,
                       float* __restrict__ cat, float* __restrict__ optraw) {
  const int t = threadIdx.x;
  const int wave = t >> 5, lane = t & 31;
  const int job = blockIdx.x*8 + wave;               // < 1152
  const int h = job / 96, rem = job % 96;
  const int it = rem / 3, nt = rem % 3;
  const int rc = lane & 15, hi = lane >> 4, koff = hi*2;
  const float* arow = &attn[((size_t)h*NRES + it*16 + rc)*NRES];
  v8f c = {};
  for (int ks = 0; ks < 128; ++ks) {
    int k0 = ks*4 + koff;
    v2f a, b;
    a.x = arow[k0]; a.y = arow[k0+1];
    b.x = vc[(h*NRES + k0)*48 + nt*16 + rc];
    b.y = vc[(h*NRES + k0 + 1)*48 + nt*16 + rc];
    c = wmma4(a, b, c);
  }
  #pragma unroll
  for (int r = 0; r < 8; ++r) {
    int m = r + hi*8;
    int ig = it*16 + m, ch = nt*16 + rc;
    if (ch < 16)      cat[(size_t)ig*OUTDIM + h*16 + ch] = c[r];
    else if (ch < 40) optraw[(ig*HN + h)*24 + (ch - 16)] = c[r];
  }
}

// ---------------------------------------------------------------------------
// K4b: o_pair[i] = a[:, i, :] @ z[i].  M=heads(pad 16), K=512, N=128 (8 tiles).
// z is served from L2 (134MB fits in 192MB).
// ---------------------------------------------------------------------------
__global__ void k4b_opair(const float* __restrict__ attn, const float* __restrict__ z,
                          float* __restrict__ cat) {
  const int t = threadIdx.x;
  const int wave = t >> 5, lane = t & 31;
  const int job = blockIdx.x*8 + wave;               // < 4096
  const int i = job >> 3, nt = job & 7;
  const int rc = lane & 15, hi = lane >> 4, koff = hi*2;
  v8f c = {};
  for (int ks = 0; ks < 128; ++ks) {
    int k0 = ks*4 + koff;
    v2f a, b;
    if (rc < HN) {
      a.x = attn[((size_t)rc*NRES + i)*NRES + k0];
      a.y = attn[((size_t)rc*NRES + i)*NRES + k0 + 1];
    } else { a.x = 0.f; a.y = 0.f; }
    b.x = z[((size_t)i*NRES + k0)*CZD + nt*16 + rc];
    b.y = z[((size_t)i*NRES + k0 + 1)*CZD + nt*16 + rc];
    c = wmma4(a, b, c);
  }
  #pragma unroll
  for (int r = 0; r < 8; ++r) {
    int m = r + hi*8;
    if (m < HN) cat[(size_t)i*OUTDIM + 576 + m*CZD + nt*16 + rc] = c[r];
  }
}

// ---------------------------------------------------------------------------
// K5a: inverse-frame rotate o_pt, norms, scatter into cat sections.
// ---------------------------------------------------------------------------
__global__ void k5a_pts(const float* __restrict__ optraw, const float* __restrict__ rot,
                        const float* __restrict__ trans, float* __restrict__ cat) {
  int idx = blockIdx.x*256 + threadIdx.x;
  if (idx >= NRES*HN*PVN) return;
  int i = idx / (HN*PVN);
  int pid = idx % (HN*PVN);
  int h = pid >> 3, p = pid & 7;
  const float* raw = &optraw[(i*HN + h)*24 + p*3];
  float y0 = raw[0] - trans[i*3+0];
  float y1 = raw[1] - trans[i*3+1];
  float y2 = raw[2] - trans[i*3+2];
  const float* R = &rot[i*9];
  float o0 = R[0]*y0 + R[3]*y1 + R[6]*y2;   // out[x] = sum_y R[y][x]*in[y]
  float o1 = R[1]*y0 + R[4]*y1 + R[7]*y2;
  float o2 = R[2]*y0 + R[5]*y1 + R[8]*y2;
  float nrm = sqrtf(o0*o0 + o1*o1 + o2*o2 + EPS_);
  float* crow = &cat[(size_t)i*OUTDIM];
  crow[192 + pid] = o0;
  crow[288 + pid] = o1;
  crow[384 + pid] = o2;
  crow[480 + pid] = nrm;
}

// ---------------------------------------------------------------------------
// K5b: out = cat(512x2112) @ Wout(2112x384) + bout.  K=2112 -> 528 WMMA steps.
// ---------------------------------------------------------------------------
__global__ void k5b_out(const float* __restrict__ cat, const float* __restrict__ Wout,
                        const float* __restrict__ bout, float* __restrict__ out) {
  const int t = threadIdx.x;
  const int wave = t >> 5, lane = t & 31;
  const int job = blockIdx.x*8 + wave;               // < 768
  const int it = job / 24, nt = job % 24;
  const int rc = lane & 15, hi = lane >> 4, koff = hi*2;
  v8f c = {};
  for (int ks = 0; ks < 528; ++ks) {
    int k0 = ks*4 + koff;
    v2f a, b;
    a.x = cat[(size_t)(it*16 + rc)*OUTDIM + k0];
    a.y = cat[(size_t)(it*16 + rc)*OUTDIM + k0 + 1];
    b.x = Wout[(size_t)k0*CSD + nt*16 + rc];
    b.y = Wout[(size_t)(k0+1)*CSD + nt*16 + rc];
    c = wmma4(a, b, c);
  }
  #pragma unroll
  for (int r = 0; r < 8; ++r) {
    int m = r + hi*8;
    out[(size_t)(it*16 + m)*CSD + nt*16 + rc] = c[r] + bout[nt*16 + rc];
  }
}

// ---------------------------------------------------------------------------
extern "C" void kernel_launch(void* const* d_in, const int* in_sizes, int n_in,
                              void* d_out, int out_size, void* d_ws, size_t ws_size,
                              hipStream_t stream) {
  const float* s      = (const float*)d_in[0];
  const float* z      = (const float*)d_in[1];
  const float* rot    = (const float*)d_in[2];
  const float* trans  = (const float*)d_in[3];
  const float* mask   = (const float*)d_in[4];
  const float* Wq     = (const float*)d_in[5];
  const float* bq     = (const float*)d_in[6];
  const float* Wkv    = (const float*)d_in[7];
  const float* bkv    = (const float*)d_in[8];
  const float* Wqp    = (const float*)d_in[9];
  const float* bqp    = (const float*)d_in[10];
  const float* Wkvp   = (const float*)d_in[11];
  const float* bkvp   = (const float*)d_in[12];
  const float* Wb     = (const float*)d_in[13];
  const float* bb     = (const float*)d_in[14];
  const float* headw  = (const float*)d_in[15];
  const float* Wout   = (const float*)d_in[16];
  const float* bout   = (const float*)d_in[17];
  float* out = (float*)d_out;

  float* ws = (float*)d_ws;
  // workspace layout (floats)
  float* qc     = ws;                         // 12*512*32  = 196608
  float* kc     = qc     + 196608;            // 196608
  float* vc     = kc     + 196608;            // 12*512*48  = 294912
  float* rb     = vc     + 294912;            // 6144
  float* cb     = rb     + 6144;              // 6144
  float* bpair  = cb     + 6144;              // 512*512*12 = 3145728
  float* attn   = bpair  + 3145728;           // 3145728
  float* optraw = attn   + 3145728;           // 512*12*24  = 147456
  float* cat    = optraw + 147456;            // 512*2112   = 1081344
  // total: 8220672 floats (~32.9 MB)

  k1_proj<<<NRES, 256, 0, stream>>>(s, rot, trans, Wq, bq, Wkv, bkv, Wqp, bqp,
                                    Wkvp, bkvp, headw, qc, kc, vc, rb, cb);
  k2_bpair<<<2048, 256, 0, stream>>>(z, Wb, bb, bpair);
  k3_attn<<<HN*32, 256, 0, stream>>>(qc, kc, rb, cb, bpair, mask, attn);
  k4a_ov<<<144, 256, 0, stream>>>(attn, vc, cat, optraw);
  k4b_opair<<<512, 256, 0, stream>>>(attn, z, cat);
  k5a_pts<<<192, 256, 0, stream>>>(optraw, rot, trans, cat);
  k5b_out<<<96, 256, 0, stream>>>(cat, Wout, bout, out);
}